// ScaledDotProductAttention_24197845746247
// MI455X (gfx1250) — compile-verified
//
#include <hip/hip_runtime.h>
#include <hip/hip_bf16.h>

typedef float v2f __attribute__((ext_vector_type(2)));
typedef float v8f __attribute__((ext_vector_type(8)));

#define SLEN 2048
#define DDIM 64
#define HNUM 16
#define LDS_STRIDE 2052  // 2048 + 4 pad -> conflict-free strided LDS access

__device__ __forceinline__ v8f wmma_f32(v2f a, v2f b, v8f c) {
  // D = A(16x4 f32) * B(4x16 f32) + C(16x16 f32)
  return __builtin_amdgcn_wmma_f32_16x16x4_f32(
      /*neg_a=*/false, a, /*neg_b=*/false, b,
      /*c_mod=*/(short)0, c, /*reuse_a=*/false, /*reuse_b=*/false);
}

__global__ void sdpa_f32_wmma_kernel(const float* __restrict__ q,
                                     const float* __restrict__ k,
                                     const float* __restrict__ v,
                                     const int*   __restrict__ mask,
                                     float* __restrict__ out,
                                     float* __restrict__ score) {
  extern __shared__ float smem[];
  float* scoreS  = smem;                       // 16 * LDS_STRIDE
  float* partS   = scoreS + 16 * LDS_STRIDE;   // 4 * 1024 PV partials
  float* redS    = partS + 4096;               // 128 reduction scratch
  float* rowmaxS = redS + 128;                 // 16
  float* rowinvS = rowmaxS + 16;               // 16

  const int qt   = blockIdx.x;       // 0..127 (q tile of 16 rows)
  const int h    = blockIdx.y;       // 0..15
  const int b    = blockIdx.z;       // 0..1
  const int tid  = threadIdx.x;      // 0..127 (4 waves)
  const int wave = tid >> 5;
  const int lane = tid & 31;
  const int mrow = lane & 15;        // row (A) / col (B,C,D) index within tile
  const int half = lane >> 4;        // which K-pair half for A/B frags

  const int q0 = qt * 16;
  const size_t bh = (size_t)(b * HNUM + h);
  const float* qbase = q + (bh * SLEN + q0) * DDIM;
  const float* kbase = k + bh * SLEN * DDIM;
  const float* vbase = v + bh * SLEN * DDIM;
  const int*   mbase = mask + ((size_t)b * SLEN + q0) * SLEN;  // mask is [B,1,S,S]

  // ---------------- Load Q fragments (A-matrix layout: lanes 0-15 K={0,1}, 16-31 K={2,3})
  v2f qa[16];
  {
    const float* qr = qbase + mrow * DDIM + 2 * half;
    #pragma unroll
    for (int i = 0; i < 16; ++i) {
      qa[i].x = qr[4 * i + 0];
      qa[i].y = qr[4 * i + 1];
    }
  }

  // ---------------- Phase 1: S = mask ? (Q K^T)/8 : -1e4  -> LDS stripe (16 x 2048)
  for (int kt = wave; kt < SLEN / 16; kt += 4) {
    const float* kr = kbase + (size_t)(kt * 16 + mrow) * DDIM + 2 * half;
    v2f kb[16];
    #pragma unroll
    for (int i = 0; i < 16; ++i) {
      kb[i].x = kr[4 * i + 0];
      kb[i].y = kr[4 * i + 1];
    }
    // Two independent accumulator chains: halves the D->C dependent latency
    v8f acc0 = {};
    v8f acc1 = {};
    #pragma unroll
    for (int i = 0; i < 16; i += 2) {
      acc0 = wmma_f32(qa[i + 0], kb[i + 0], acc0);
      acc1 = wmma_f32(qa[i + 1], kb[i + 1], acc1);
    }
    const v8f acc = acc0 + acc1;

    const int col = kt * 16 + mrow;  // C/D: n = lane&15
    #pragma unroll
    for (int r = 0; r < 8; ++r) {
      const int m = r + 8 * half;    // C/D: m = vgpr + 8*(lane>=16)
      const int mv = mbase[(size_t)m * SLEN + col];
      float s = acc[r] * 0.125f;     // 1/sqrt(64)
      s = (mv == 0) ? -10000.0f : s;
      scoreS[m * LDS_STRIDE + col] = s;
    }
  }
  __syncthreads();

  // ---------------- Phase 2: softmax statistics (row max, exp, row sum)
  {
    const int row = tid & 15;
    const int chunk = tid >> 4;  // 8 chunks of 256 columns
    const float* rp = scoreS + row * LDS_STRIDE + chunk * 256;
    float mx = -3.0e38f;
    for (int i = 0; i < 256; ++i) mx = fmaxf(mx, rp[i]);
    redS[tid] = mx;
  }
  __syncthreads();
  if (tid < 16) {
    float mx = redS[tid];
    #pragma unroll
    for (int c = 1; c < 8; ++c) mx = fmaxf(mx, redS[tid + 16 * c]);
    rowmaxS[tid] = mx;
  }
  __syncthreads();
  {
    const int row = tid & 15;
    const int chunk = tid >> 4;
    float* rp = scoreS + row * LDS_STRIDE + chunk * 256;
    const float mx = rowmaxS[row];
    float sum = 0.0f;
    for (int i = 0; i < 256; ++i) {
      const float e = __expf(rp[i] - mx);
      rp[i] = e;            // keep exp in LDS
      sum += e;
    }
    redS[tid] = sum;
  }
  __syncthreads();
  if (tid < 16) {
    float s = 0.0f;
    #pragma unroll
    for (int c = 0; c < 8; ++c) s += redS[tid + 16 * c];
    rowinvS[tid] = 1.0f / s;
  }
  __syncthreads();

  // ---------------- Phase 3: normalize; stream score to HBM (nontemporal)
  for (int m = 0; m < 16; ++m) {
    const float inv = rowinvS[m];
    float* rp = scoreS + m * LDS_STRIDE;
    float* gp = score + (bh * SLEN + (size_t)(q0 + m)) * SLEN;
    for (int cb = 0; cb < SLEN; cb += 128) {
      const float p = rp[cb + tid] * inv;
      rp[cb + tid] = p;  // normalized prob kept in LDS for PV
      __builtin_nontemporal_store(p, gp + cb + tid);
    }
  }
  __syncthreads();

  // ---------------- Phase 4: out = P (16x2048) x V (2048x64), waves split K range
  v8f zero = {};
  v8f oacc[4];  // 4 independent accumulator chains (one per 16-col N tile)
  #pragma unroll
  for (int nt = 0; nt < 4; ++nt) oacc[nt] = zero;

  const int kd0 = wave * (SLEN / 4);
  for (int kd = kd0; kd < kd0 + SLEN / 4; kd += 4) {
    v2f a;
    a.x = scoreS[mrow * LDS_STRIDE + kd + 2 * half + 0];
    a.y = scoreS[mrow * LDS_STRIDE + kd + 2 * half + 1];
    #pragma unroll
    for (int nt = 0; nt < 4; ++nt) {
      const float* vp = vbase + (size_t)(kd + 2 * half) * DDIM + nt * 16 + mrow;
      v2f bf;
      bf.x = vp[0];
      bf.y = vp[DDIM];
      oacc[nt] = wmma_f32(a, bf, oacc[nt]);
    }
  }
  #pragma unroll
  for (int nt = 0; nt < 4; ++nt)
    #pragma unroll
    for (int r = 0; r < 8; ++r)
      partS[wave * 1024 + nt * 256 + r * 32 + lane] = oacc[nt][r];
  __syncthreads();

  // cross-wave reduce + coalesced write of out tile (16 x 64)
  for (int i = tid; i < 1024; i += 128) {
    const float s = partS[i] + partS[i + 1024] + partS[i + 2048] + partS[i + 3072];
    const int nt  = i >> 8;
    const int rem = i & 255;
    const int r   = rem >> 5;
    const int ln  = rem & 31;
    const int m   = r + 8 * (ln >> 4);
    const int n   = nt * 16 + (ln & 15);
    __builtin_nontemporal_store(s, out + (bh * SLEN + (size_t)(q0 + m)) * DDIM + n);
  }
}

extern "C" void kernel_launch(void* const* d_in, const int* in_sizes, int n_in,
                              void* d_out, int out_size, void* d_ws, size_t ws_size,
                              hipStream_t stream) {
  const float* q    = (const float*)d_in[0];
  const float* k    = (const float*)d_in[1];
  const float* v    = (const float*)d_in[2];
  const int*   mask = (const int*)d_in[3];

  float* out   = (float*)d_out;                          // [B,H,S,D] first
  float* score = out + (size_t)2 * HNUM * SLEN * DDIM;   // then [B,H,S,S]

  const size_t lds_bytes =
      (size_t)(16 * LDS_STRIDE + 4096 + 128 + 32) * sizeof(float);  // ~145 KB

  hipFuncSetAttribute((const void*)sdpa_f32_wmma_kernel,
                      hipFuncAttributeMaxDynamicSharedMemorySize, (int)lds_bytes);

  dim3 grid(SLEN / 16, HNUM, 2);  // (q tile, h, b) = 4096 blocks
  sdpa_f32_wmma_kernel<<<grid, 128, lds_bytes, stream>>>(q, k, v, mask, out, score);
}